// Hash1d_59906203844970
// MI455X (gfx1250) — compile-verified
//
#include <hip/hip_runtime.h>

#define INPUT_DIM 16384
#define EMB_SIZE  1024
#define BATCH     4096

typedef unsigned int v4u __attribute__((ext_vector_type(4)));
typedef int          v4i __attribute__((ext_vector_type(4)));
typedef int          v8i __attribute__((ext_vector_type(8)));
typedef float        v4f __attribute__((ext_vector_type(4)));
typedef unsigned int v2u __attribute__((ext_vector_type(2)));

// ---------------------------------------------------------------------------
// Kernel 1: recover (bucket, sign) per input feature from the dense hashProj.
// One wave32 per row; each lane scans 8 float4's (coalesced 512B/wave/iter),
// non-temporal (hashProj is read exactly once; don't pollute L2).
// Exactly one element per row is nonzero, so exactly one predicated store fires.
// code = bucket[9:0] | (sign<0 ? 0x8000 : 0), stored as u16 (32 KB total).
// ---------------------------------------------------------------------------
__global__ void __launch_bounds__(256)
hash_extract_kernel(const float* __restrict__ hashProj,
                    unsigned short* __restrict__ codes) {
  const int row  = blockIdx.x * 8 + (threadIdx.x >> 5);
  const int lane = threadIdx.x & 31;
  const v4f* r = (const v4f*)(hashProj + (size_t)row * EMB_SIZE);
#pragma unroll
  for (int k = 0; k < 8; ++k) {
    const int i4 = lane + 32 * k;          // float4 index, coalesced
    const v4f v = __builtin_nontemporal_load(r + i4);
    const unsigned base = (unsigned)i4 * 4u;
    if (v.x != 0.0f) codes[row] = (unsigned short)((base    ) | (v.x < 0.0f ? 0x8000u : 0u));
    if (v.y != 0.0f) codes[row] = (unsigned short)((base + 1) | (v.y < 0.0f ? 0x8000u : 0u));
    if (v.z != 0.0f) codes[row] = (unsigned short)((base + 2) | (v.z < 0.0f ? 0x8000u : 0u));
    if (v.w != 0.0f) codes[row] = (unsigned short)((base + 3) | (v.w < 0.0f ? 0x8000u : 0u));
  }
}

// ---------------------------------------------------------------------------
// Kernel 2: one block per batch row. TDM async-copies the 32KB code table
// into LDS (overlapped with zeroing the 4KB accumulator), then streams the
// 64KB x-row with non-temporal global_load_b128 and scatter-adds via
// ds_add_f32. The code table stays RT-cached in L2 (hot across 4096 blocks);
// x/out are single-touch and bypass with TH=NT.
// ---------------------------------------------------------------------------
struct Smem {
  unsigned short codes[INPUT_DIM];  // 32 KB, filled by tensor_load_to_lds
  float          acc[EMB_SIZE];     // 4 KB accumulator
};

__global__ void __launch_bounds__(512)
hash_scatter_kernel(const float* __restrict__ x,
                    const unsigned short* __restrict__ codes_g,
                    float* __restrict__ out) {
  __shared__ Smem s;
  const int tid = threadIdx.x;
  const int b   = blockIdx.x;

  // ---- Wave 0 issues the TDM descriptor (CDNA5 ISA) ----
  if (tid < 32) {
    // Flat address of LDS location: aperture in bits[63:32], offset in [31:0].
    const unsigned lds = (unsigned)(unsigned long long)(&s.codes[0]);
    const unsigned long long ga = (unsigned long long)codes_g;

    v4u g0 = {
      1u,                                                    // count=1 valid descriptor
      lds,                                                   // lds_addr (bytes)
      (unsigned)(ga & 0xffffffffull),                        // global_addr[31:0]
      (unsigned)((ga >> 32) & 0x1ffffffull) | (2u << 30)     // global_addr[56:32] | type=2
    };
    v8i g1 = {
      (int)(1u << 16),                                       // data_size=1 -> 2 bytes/elem
      (int)((INPUT_DIM & 0xffffu) << 16),                    // tensor_dim0[15:0] in [31:16]
      (int)(((INPUT_DIM >> 16) & 0xffffu) | (1u << 16)),     // tensor_dim0 hi | tensor_dim1=1
      (int)((INPUT_DIM & 0xffffu) << 16),                    // tile_dim0 = 16384
      1,                                                     // tile_dim1=1, tile_dim2=0
      INPUT_DIM,                                             // tensor_dim0_stride lo
      0, 0                                                   // stride hi, tensor_dim1_stride
    };
    v4i g2 = {0, 0, 0, 0};                                   // <=2D: groups 2/3 unused
    v4i g3 = {0, 0, 0, 0};
    v8i g4 = {0, 0, 0, 0, 0, 0, 0, 0};                       // clang-23 6-arg form: extra
                                                             // group, zero-filled
    __builtin_amdgcn_tensor_load_to_lds(g0, g1, g2, g3, g4, 0);
  }

  // Zero the accumulator while the DMA is in flight.
  for (int e = tid; e < EMB_SIZE; e += 512) s.acc[e] = 0.0f;

  if (tid < 32) __builtin_amdgcn_s_wait_tensorcnt(0);        // TENSORcnt==0 -> LDS ready
  __syncthreads();

  // ---- Streaming scatter: 8 iterations of NT b128 load + ds_load_b64 + 4 ds_add_f32 ----
  const v4f* xrow = (const v4f*)(x + (size_t)b * INPUT_DIM);
  const v2u* c2   = (const v2u*)s.codes;
#pragma unroll
  for (int kk = 0; kk < INPUT_DIM / 4 / 512; ++kk) {
    const int k = tid + kk * 512;
    const v4f v  = __builtin_nontemporal_load(xrow + k);
    const v2u cc = c2[k];
    const unsigned c0 = cc.x & 0xffffu, c1 = cc.x >> 16;
    const unsigned c2v = cc.y & 0xffffu, c3 = cc.y >> 16;
    atomicAdd(&s.acc[c0  & (EMB_SIZE - 1)], (c0  & 0x8000u) ? -v.x : v.x);
    atomicAdd(&s.acc[c1  & (EMB_SIZE - 1)], (c1  & 0x8000u) ? -v.y : v.y);
    atomicAdd(&s.acc[c2v & (EMB_SIZE - 1)], (c2v & 0x8000u) ? -v.z : v.z);
    atomicAdd(&s.acc[c3  & (EMB_SIZE - 1)], (c3  & 0x8000u) ? -v.w : v.w);
  }
  __syncthreads();

  // ---- Write the finished 4KB row with NT b128 stores (single-touch output) ----
  v4f* orow = (v4f*)(out + (size_t)b * EMB_SIZE);
  for (int q = tid; q < EMB_SIZE / 4; q += 512) {
    v4f r;
    r.x = s.acc[4 * q + 0];
    r.y = s.acc[4 * q + 1];
    r.z = s.acc[4 * q + 2];
    r.w = s.acc[4 * q + 3];
    __builtin_nontemporal_store(r, orow + q);
  }
}

// ---------------------------------------------------------------------------
extern "C" void kernel_launch(void* const* d_in, const int* in_sizes, int n_in,
                              void* d_out, int out_size, void* d_ws, size_t ws_size,
                              hipStream_t stream) {
  const float* x  = (const float*)d_in[0];   // [BATCH, INPUT_DIM] f32
  const float* hp = (const float*)d_in[1];   // [INPUT_DIM, EMB_SIZE] f32 (1 nz/row)
  float* out = (float*)d_out;                // [BATCH, EMB_SIZE] f32
  unsigned short* codes = (unsigned short*)d_ws;  // 32 KB scratch

  hash_extract_kernel<<<INPUT_DIM / 8, 256, 0, stream>>>(hp, codes);
  hash_scatter_kernel<<<BATCH, 512, 0, stream>>>(x, codes, out);
}